// EvalEig_42838003810958
// MI455X (gfx1250) — compile-verified
//
#include <hip/hip_runtime.h>

// ---------------------------------------------------------------------------
// EvalEig, MI455X (gfx1250, wave32) — wave-parallel prefix-matrix formulation.
//
// One wave32 per (energy, l) chain: 4096*3 = 12288 waves.  The backward
// Numerov recurrence is a linear map on the 4-window, so the 996 steps are
// split into 32 lane-segments of 32 steps:
//   Phase A: lane j builds its 4x4 window-transfer matrix T_j (companion prod)
//   Phase B: Kogge-Stone inclusive scan S_j = T_j * ... * T_0; each level's
//            32 independent 4x4*4x4 products are done 4-at-a-time as the
//            diagonal blocks of V_WMMA_F32_16X16X4_F32 (8 wmma/level, 5
//            levels), marshaled through per-wave LDS.  LDS is in-order per
//            wave on CDNA5 (DScnt), so cross-lane exchange needs only a
//            *compiler* fence: __builtin_amdgcn_wave_barrier() (0 instr).
//            D-writeback selects the *address* (real slot vs dump row), never
//            the value, so all vector extracts stay constant-index.
//   Phase C: lane j applies S_{j-1} to the initial window -> segment start.
//   Phase D: lane j replays its 32 steps with the faithful scalar step to
//            accumulate the integral (f2 term unsquared, reference quirk) and
//            capture v[899..903]; shfl reductions finish.
// Forward chain only needs k<=100 -> all lanes compute it redundantly
// (wave-uniform => free on SIMD32).  f32 throughout: the recurrence is
// marginally stable over 1000 steps; low-precision WMMA types would destroy
// it.  16x16x4 f32 is the CDNA5 matrix shape that fits 4x4 block products.
// ---------------------------------------------------------------------------

typedef float v2f __attribute__((ext_vector_type(2)));
typedef float v8f __attribute__((ext_vector_type(8)));

#if defined(__has_builtin)
#if __has_builtin(__builtin_amdgcn_wmma_f32_16x16x4_f32)
#define EVE_HAVE_WMMA 1
#endif
#endif
#ifndef EVE_HAVE_WMMA
#define EVE_HAVE_WMMA 0
#endif

__device__ __forceinline__ void wave_sync_lds() {
#if defined(__has_builtin)
#if __has_builtin(__builtin_amdgcn_wave_barrier)
    __builtin_amdgcn_wave_barrier();   // compiler-only fence; LDS is in-order
#else
    __threadfence_block();
#endif
#else
    __threadfence_block();
#endif
}

namespace {

constexpr int kRN = 1000;
constexpr int kNL = 3;
constexpr int kWavesPerBlock = 8;
constexpr int kBlockDim = kWavesPerBlock * 32;

constexpr double kRDd  = 100.0 / 1000.0;
constexpr double kRD2d = kRDd * kRDd;

constexpr float kCA = (float)(13.0 / 15.0 * kRD2d);   // a = 2 + kCA*f
constexpr float kCB = (float)(7.0 / 60.0 * kRD2d);    // b = -2 + kCB*f
constexpr float kCC = (float)(3.0 / 40.0 * kRD2d);    // c = -1+kCC*f ; d = 1-kCC*f
constexpr float kR0   = (float)(100.0 / 1000.0);
constexpr float kStep = (float)((100.0 - 100.0 / 1000.0) / 999.0);
constexpr float kIntegScale = (float)(2.0 * kRDd / 45.0);
constexpr float kDerivDen   = (float)(12.0 * kRDd);

__device__ __forceinline__ float r_of(int i) {
    return fmaf((float)i, kStep, kR0);
}

// Faithful integrate_sq accumulation (last = 4n). idx%4==2 term is UNSQUARED.
__device__ __forceinline__ void integ_add(float& acc, float v, int idx, int last) {
    if ((unsigned)idx > (unsigned)last) return;
    int m = idx & 3;
    if (m == 2) {
        acc += 12.0f * v;
    } else if (m == 0) {
        float w = ((idx == 0) || (idx == last)) ? 7.0f : 14.0f;
        acc += w * v * v;
    } else {
        acc += 32.0f * v * v;
    }
}

__global__ __launch_bounds__(kBlockDim)
void eval_eig_scan_kernel(const float* __restrict__ energy,
                          float* __restrict__ out,
                          int nE) {
    // {1/r, 1/r^2} grid table, shared by the 8 waves of the block.
    __shared__ float2 tab[kRN];
    // Per-wave scan scratch: slots 0..31 = S matrices, slot 32 = identity.
    // Double buffered (Kogge-Stone levels must read old values).
    __shared__ __align__(16) float mats[2][kWavesPerBlock][33][16];
    // Write-only dump row for unwanted (off-diagonal) WMMA block elements.
    __shared__ float dump[kWavesPerBlock][16];

    for (int i = threadIdx.x; i < kRN; i += blockDim.x) {
        float r = r_of(i);
        tab[i] = make_float2(1.0f / r, 1.0f / (r * r));
    }
    __syncthreads();

    const int lane  = threadIdx.x & 31;
    const int wslot = threadIdx.x >> 5;
    const int chain = blockIdx.x * kWavesPerBlock + wslot;   // one wave per chain
    const int total = nE * kNL;
    if (chain >= total) return;                              // wave-uniform exit

    const int ie = chain / kNL;
    const int il = chain - ie * kNL;

    __builtin_prefetch(energy + ie, 0, 0);                   // global_prefetch_b8
    const float e   = energy[ie];
    const float fl  = (float)il;
    const float ll1 = fl * (fl + 1.0f);

    auto factorTab = [&](int i) -> float {                   // factor[i]
        float2 t = tab[i];
        return fmaf(ll1, t.y, -e - t.x);
    };

    // Identity matrix into slot 32 of both buffers (lanes 0..15, one elem each).
    if (lane < 16) {
        float idv = ((lane >> 2) == (lane & 3)) ? 1.0f : 0.0f;
        mats[0][wslot][32][lane] = idv;
        mats[1][wslot][32][lane] = idv;
    }

    // ================= forward chain (k = 0..100), all lanes redundantly ====
    float lf_in, integ_in, u100;
    {
        float s0, s1, s2, s3;
        const float den = 2.0f * (fl + 1.0f);
        {
            float uu[4];
#pragma unroll
            for (int i = 0; i < 4; ++i) {
                float r  = r_of(i);
                float p1 = r;
                for (int q = 0; q < il; ++q) p1 *= r;        // r^(l+1)
                uu[i] = p1 - (p1 * r) / den;
            }
            s0 = uu[0]; s1 = uu[1]; s2 = uu[2]; s3 = uu[3];
        }
        float fm4 = factorTab(0), fm3 = factorTab(1),
              fm2 = factorTab(2), fm1 = factorTab(3);
        float acc_in = 0.0f;
        integ_add(acc_in, s0, 0, 96);
        integ_add(acc_in, s1, 1, 96);
        integ_add(acc_in, s2, 2, 96);
        integ_add(acc_in, s3, 3, 96);
        float u96 = 0.0f;
#pragma unroll 4
        for (int k = 4; k <= 100; ++k) {
            float fk = factorTab(k);
            float dk = fmaf(-kCC, fk, 1.0f);
            float neu = s3 * fmaf(kCA, fm1, 2.0f)
                      + s2 * fmaf(kCB, fm2, -2.0f)
                      + s1 * fmaf(kCA, fm3, 2.0f)
                      + (s0 * fmaf(kCC, fm4, -1.0f)) / dk;
            integ_add(acc_in, neu, k, 96);
            if (k == 96) u96 = neu;
            s0 = s1; s1 = s2; s2 = s3; s3 = neu;
            fm4 = fm3; fm3 = fm2; fm2 = fm1; fm1 = fk;
        }
        u100 = s3;
        lf_in = ((25.0f * s3 - 48.0f * s2 + 36.0f * s1 - 16.0f * s0 + 3.0f * u96)
                 / kDerivDen) / u100;
        integ_in = acc_in * kIntegScale;
    }

    // =========== backward chain initial window W0 = ui_init[0..3] ==========
    float w0i, w1i, w2i, w3i;
    {
        const float se    = sqrtf(fabsf(e));
        const float factl = (il == 0) ? 1.0f : ((il == 1) ? 3.0f : 15.0f);
        const float d1    = 2.0f * fl + 3.0f;
        const float d2    = 2.0f * d1 * (2.0f * fl + 5.0f);
        float uu[4];
#pragma unroll
        for (int i = 0; i < 4; ++i) {
            float r = r_of(i);
            float x = r * se;
            float base = 1.0f;
            for (int q = 0; q < il; ++q) base *= x;          // x^l
            base = base / factl;
            float h = x * x / 2.0f;
            uu[i] = r * base * (1.0f + h / d1 + (h * h) / d2);
        }
        w0i = uu[0]; w1i = uu[1]; w2i = uu[2]; w3i = uu[3];
    }

    // ============ Phase A: per-lane segment transfer matrix T_j ============
    // Lane j covers steps k in [32j+4, 32j+35]; k > 999 are identity padding.
    const int k0 = 32 * lane + 4;
    float P[16];
#pragma unroll
    for (int r = 0; r < 16; ++r) P[r] = ((r >> 2) == (r & 3)) ? 1.0f : 0.0f;
    {
        float gm4 = factorTab(999 - (k0 - 4));
        float gm3 = factorTab(999 - (k0 - 3));
        float gm2 = factorTab(999 - (k0 - 2));
        float gm1 = factorTab(999 - (k0 - 1));
#pragma unroll 4
        for (int s = 0; s < 32; ++s) {
            const int k = k0 + s;
            if (k <= 999) {
                float gk = factorTab(999 - k);
                float a3 = fmaf(kCA, gm1, 2.0f);
                float b2 = fmaf(kCB, gm2, -2.0f);
                float a1 = fmaf(kCA, gm3, 2.0f);
                float cd = fmaf(kCC, gm4, -1.0f) / fmaf(-kCC, gk, 1.0f);
                // companion multiply: rows shift up, new row3 = combo
#pragma unroll
                for (int c = 0; c < 4; ++c) {
                    float nr = fmaf(cd, P[c],
                               fmaf(a1, P[4 + c],
                               fmaf(b2, P[8 + c], a3 * P[12 + c])));
                    P[c] = P[4 + c]; P[4 + c] = P[8 + c];
                    P[8 + c] = P[12 + c]; P[12 + c] = nr;
                }
                gm4 = gm3; gm3 = gm2; gm2 = gm1; gm1 = gk;
            }
        }
    }
    {
        float4* mv = reinterpret_cast<float4*>(&mats[0][wslot][lane][0]);
        mv[0] = make_float4(P[0],  P[1],  P[2],  P[3]);
        mv[1] = make_float4(P[4],  P[5],  P[6],  P[7]);
        mv[2] = make_float4(P[8],  P[9],  P[10], P[11]);
        mv[3] = make_float4(P[12], P[13], P[14], P[15]);
    }
    wave_sync_lds();   // other lanes' T_j / identity now consumable

    // ======= Phase B: Kogge-Stone scan of S_j = T_j * ... * T_0 ===========
    int src = 0;
#pragma unroll
    for (int lvl = 0; lvl < 5; ++lvl) {
        const int dist = 1 << lvl;
        const int dst  = 1 - src;
        const int Lp      = lane & 15;
        const int topHalf = lane >> 4;       // 0: K/rows {0,1}; 1: K/rows {2,3}
        const int tIdx    = Lp >> 2;         // which matrix of the group of 4
        const int pq      = Lp & 3;          // A row / B-D column within block
#if EVE_HAVE_WMMA
        // 8 WMMAs cover 32 matrices: group g packs left mats 4g..4g+3 as A
        // row-blocks and right mats as B col-blocks; diagonal D blocks are
        // the wanted products.
#pragma unroll
        for (int g = 0; g < 8; ++g) {
            const int jA = 4 * g + tIdx;
            const int jB = (jA >= dist) ? (jA - dist) : 32;   // 32 = identity
            const float* MA = &mats[src][wslot][jA][0];
            const float* MB = &mats[src][wslot][jB][0];
            // A 16x4 layout: row = Lp, VGPR0/1 = K (2*topHalf, 2*topHalf+1)
            v2f av = *reinterpret_cast<const v2f*>(MA + pq * 4 + 2 * topHalf);
            // B 4x16 layout (mirrors C/D row striping): col = Lp,
            // VGPR0/1 = K rows (2*topHalf, 2*topHalf+1)
            v2f bv;
            bv.x = MB[(2 * topHalf) * 4 + pq];
            bv.y = MB[(2 * topHalf + 1) * 4 + pq];
            v8f cz = {};
            v8f dres = __builtin_amdgcn_wmma_f32_16x16x4_f32(
                false, av, false, bv, (short)0, cz, false, false);
            // D layout: this lane holds column pq of row-blocks; its
            // dres[0..3] is abs rows 8*topHalf+0..3 (block 2*topHalf) and
            // dres[4..7] is abs rows 8*topHalf+4..7 (block 2*topHalf+1).
            // The block is wanted iff its row-block == tIdx; select the
            // ADDRESS (real slot vs dump), never the value, so all vector
            // extracts keep constant indices.
            const bool loSel = (tIdx == 2 * topHalf);
            const bool hiSel = (tIdx == 2 * topHalf + 1);
            float* MD  = &mats[dst][wslot][jA][0];
            float* dmp = &dump[wslot][0];
            float* tLo = loSel ? MD : dmp;
            float* tHi = hiSel ? MD : dmp;
            tLo[0 * 4 + pq] = dres[0];
            tLo[1 * 4 + pq] = dres[1];
            tLo[2 * 4 + pq] = dres[2];
            tLo[3 * 4 + pq] = dres[3];
            tHi[0 * 4 + pq] = dres[4];
            tHi[1 * 4 + pq] = dres[5];
            tHi[2 * 4 + pq] = dres[6];
            tHi[3 * 4 + pq] = dres[7];
        }
#else
        // FMA fallback: per-lane 4x4 product through LDS.
        {
            const int jB = (lane >= dist) ? (lane - dist) : 32;
            const float4* MLq = reinterpret_cast<const float4*>(
                &mats[src][wslot][lane][0]);
            const float4* MRq = reinterpret_cast<const float4*>(
                &mats[src][wslot][jB][0]);
            float4 L0 = MLq[0], L1 = MLq[1], L2 = MLq[2], L3 = MLq[3];
            float4 R0 = MRq[0], R1 = MRq[1], R2 = MRq[2], R3 = MRq[3];
            float Lm[16] = {L0.x,L0.y,L0.z,L0.w, L1.x,L1.y,L1.z,L1.w,
                            L2.x,L2.y,L2.z,L2.w, L3.x,L3.y,L3.z,L3.w};
            float Rm[16] = {R0.x,R0.y,R0.z,R0.w, R1.x,R1.y,R1.z,R1.w,
                            R2.x,R2.y,R2.z,R2.w, R3.x,R3.y,R3.z,R3.w};
            float* MD = &mats[dst][wslot][lane][0];
#pragma unroll
            for (int rr = 0; rr < 4; ++rr) {
#pragma unroll
                for (int cc = 0; cc < 4; ++cc) {
                    float a2 = Lm[rr * 4 + 0] * Rm[0 * 4 + cc];
                    a2 = fmaf(Lm[rr * 4 + 1], Rm[1 * 4 + cc], a2);
                    a2 = fmaf(Lm[rr * 4 + 2], Rm[2 * 4 + cc], a2);
                    a2 = fmaf(Lm[rr * 4 + 3], Rm[3 * 4 + cc], a2);
                    MD[rr * 4 + cc] = a2;
                }
            }
        }
        (void)topHalf; (void)tIdx; (void)pq;
#endif
        wave_sync_lds();   // level results visible before next level's loads
        src = dst;
    }

    // ===== Phase C: segment start window = S_{lane-1} * W0 (I for lane 0) ==
    float s0, s1, s2, s3;
    {
        const int jprev = (lane == 0) ? 32 : (lane - 1);
        const float4* SFq = reinterpret_cast<const float4*>(
            &mats[src][wslot][jprev][0]);
        float4 r0 = SFq[0], r1 = SFq[1], r2 = SFq[2], r3 = SFq[3];
        s0 = r0.x * w0i + r0.y * w1i + r0.z * w2i + r0.w * w3i;
        s1 = r1.x * w0i + r1.y * w1i + r1.z * w2i + r1.w * w3i;
        s2 = r2.x * w0i + r2.y * w1i + r2.z * w2i + r2.w * w3i;
        s3 = r3.x * w0i + r3.y * w1i + r3.z * w2i + r3.w * w3i;
    }

    // ===== Phase D: faithful scalar replay of the 32 segment steps =========
    float acc_out = 0.0f;
    float cap899 = 0.0f, cap900 = 0.0f, cap901 = 0.0f,
          cap902 = 0.0f, cap903 = 0.0f;
    {
        float gm4 = factorTab(999 - (k0 - 4));
        float gm3 = factorTab(999 - (k0 - 3));
        float gm2 = factorTab(999 - (k0 - 2));
        float gm1 = factorTab(999 - (k0 - 1));
#pragma unroll 4
        for (int s = 0; s < 32; ++s) {
            const int k = k0 + s;
            if (k <= 999) {
                float gk = factorTab(999 - k);
                float dk = fmaf(-kCC, gk, 1.0f);
                float neu = s3 * fmaf(kCA, gm1, 2.0f)
                          + s2 * fmaf(kCB, gm2, -2.0f)
                          + s1 * fmaf(kCA, gm3, 2.0f)
                          + (s0 * fmaf(kCC, gm4, -1.0f)) / dk;
                integ_add(acc_out, neu, 999 - k, 896);   // u_infty idx j=999-k
                if      (k == 899) cap899 = neu;         // u_infty[100] (lane 27)
                else if (k == 900) cap900 = neu;         // u_infty[99]  (lane 28)
                else if (k == 901) cap901 = neu;
                else if (k == 902) cap902 = neu;
                else if (k == 903) cap903 = neu;
                s0 = s1; s1 = s2; s2 = s3; s3 = neu;
                gm4 = gm3; gm3 = gm2; gm2 = gm1; gm1 = gk;
            }
        }
    }
    // wave reductions / broadcasts
#pragma unroll
    for (int m = 16; m >= 1; m >>= 1) acc_out += __shfl_xor(acc_out, m, 32);
    const float c899 = __shfl(cap899, 27, 32);
    const float c900 = __shfl(cap900, 28, 32);
    const float c901 = __shfl(cap901, 28, 32);
    const float c902 = __shfl(cap902, 28, 32);
    const float c903 = __shfl(cap903, 28, 32);

    if (lane == 0) {
        const float lf_out =
            ((25.0f * c899 - 48.0f * c900 + 36.0f * c901 - 16.0f * c902
              + 3.0f * c903) / kDerivDen) / c899;
        const float integ_out = acc_out * kIntegScale;
        const float denom = integ_in / (u100 * u100)
                          + integ_out / (c899 * c899);
        out[chain] = e + (-(lf_out - lf_in) / denom);
    }
}

} // anonymous namespace

extern "C" void kernel_launch(void* const* d_in, const int* in_sizes, int n_in,
                              void* d_out, int out_size, void* d_ws, size_t ws_size,
                              hipStream_t stream) {
    (void)n_in; (void)out_size; (void)d_ws; (void)ws_size;
    const float* energy = (const float*)d_in[0];
    float* out = (float*)d_out;
    const int nE    = in_sizes[0];                 // 4096
    const int waves = nE * kNL;                    // one wave32 per (e,l) chain
    const int grid  = (waves + kWavesPerBlock - 1) / kWavesPerBlock;
    hipLaunchKernelGGL(eval_eig_scan_kernel, dim3(grid), dim3(kBlockDim), 0,
                       stream, energy, out, nE);
}